// PWRSWtL_6141803233717
// MI455X (gfx1250) — compile-verified
//
#include <hip/hip_runtime.h>
#include <hip/hip_bf16.h>
#include <math.h>
#include <stdint.h>

// ---------------------------------------------------------------------------
// Weighted (histogram-equalized) MSE loss, MI455X / gfx1250.
// Memory-bound: ~400 MB traffic -> ~17us floor at 23.3 TB/s.
// CDNA5 paths: global_load_async_to_lds_b128 + s_wait_asynccnt with a
// 4-stage per-wave pipeline (3 KB in flight per wave -> ~24 MB device-wide,
// above the ~16 MB Little's-law threshold), LDS-privatized histogram
// (ds atomics), wave32 shuffles, double accumulation.
// ---------------------------------------------------------------------------

#define MAXBINS 512
#define HBLOCKS 1024
#define WBLOCKS 1024
#define NTHREADS 256
#define STAGES 4            // async pipeline depth (per-wave)

// ---- float <-> order-preserving unsigned key (for atomicMax on floats) ----
__device__ __forceinline__ unsigned fkey(float f) {
    unsigned b = __float_as_uint(f);
    return (b & 0x80000000u) ? ~b : (b | 0x80000000u);
}
__device__ __forceinline__ float funkey(unsigned k) {
    unsigned b = (k & 0x80000000u) ? (k & 0x7FFFFFFFu) : ~k;
    return __uint_as_float(b);
}

// ---- CDNA5 async global->LDS (ASYNCcnt) ----------------------------------
// saddr form: mem_addr = SGPR64 + VGPR32 + inst_offset ; dest = per-lane LDS
// byte address in the VDST VGPR. Tracked by ASYNCcnt, loads complete in order.
__device__ __forceinline__ void async_ld16(uint32_t lds_byte, uint32_t goff, uint64_t base) {
    asm volatile("global_load_async_to_lds_b128 %0, %1, %2 offset:0"
                 :: "v"(lds_byte), "v"(goff), "s"(base) : "memory");
}
// allow = number of async loads permitted to remain outstanding
__device__ __forceinline__ void async_wait_allow(int allow) {
    switch (allow) {
        case 6:  asm volatile("s_wait_asynccnt 0x6" ::: "memory"); break;
        case 4:  asm volatile("s_wait_asynccnt 0x4" ::: "memory"); break;
        case 2:  asm volatile("s_wait_asynccnt 0x2" ::: "memory"); break;
        default: asm volatile("s_wait_asynccnt 0x0" ::: "memory"); break;
    }
}

// ---------------------------------------------------------------------------
// Kernel 0: init workspace
// ---------------------------------------------------------------------------
__global__ void pwl_init_kernel(unsigned* __restrict__ ghist, unsigned* __restrict__ gmaxkey) {
    int tid = threadIdx.x;           // 512 threads
    if (tid < MAXBINS) ghist[tid] = 0u;
    if (tid == 0) *gmaxkey = 0u;     // key 0 == below every real float
}

// ---------------------------------------------------------------------------
// Kernel 1: histogram of floor(tar) + max(tar). LDS-privatized, one global
// merge per block. 4-way batched b128 loads for MLP.
// ---------------------------------------------------------------------------
__device__ __forceinline__ void hist_acc(float4 v, float& m, unsigned* lh) {
    m = fmaxf(m, fmaxf(fmaxf(v.x, v.y), fmaxf(v.z, v.w)));
    int b;
    b = (int)floorf(v.x); if ((unsigned)b < (unsigned)MAXBINS) atomicAdd(&lh[b], 1u);
    b = (int)floorf(v.y); if ((unsigned)b < (unsigned)MAXBINS) atomicAdd(&lh[b], 1u);
    b = (int)floorf(v.z); if ((unsigned)b < (unsigned)MAXBINS) atomicAdd(&lh[b], 1u);
    b = (int)floorf(v.w); if ((unsigned)b < (unsigned)MAXBINS) atomicAdd(&lh[b], 1u);
}

__global__ void pwl_hist_kernel(const float* __restrict__ tar, long long n,
                                unsigned* __restrict__ ghist, unsigned* __restrict__ gmaxkey) {
    __shared__ unsigned lh[MAXBINS];
    __shared__ unsigned lmax;
    const int tid = threadIdx.x;
    for (int i = tid; i < MAXBINS; i += NTHREADS) lh[i] = 0u;
    if (tid == 0) lmax = 0u;
    __syncthreads();

    float m = -INFINITY;
    const long long nvec   = n >> 2;
    const long long stride = (long long)gridDim.x * NTHREADS;
    const float4* t4 = (const float4*)tar;

    long long i = (long long)blockIdx.x * NTHREADS + tid;
    // 4-way batched: 4 independent b128 loads in flight per thread
    for (; i + 3 * stride < nvec; i += 4 * stride) {
        float4 a = t4[i];
        float4 b = t4[i + stride];
        float4 c = t4[i + 2 * stride];
        float4 d = t4[i + 3 * stride];
        hist_acc(a, m, lh);
        hist_acc(b, m, lh);
        hist_acc(c, m, lh);
        hist_acc(d, m, lh);
    }
    for (; i < nvec; i += stride) hist_acc(t4[i], m, lh);

    // scalar tail (n % 4)
    for (long long j = (nvec << 2) + (long long)blockIdx.x * NTHREADS + tid; j < n; j += stride) {
        float v = tar[j];
        m = fmaxf(m, v);
        int b = (int)floorf(v);
        if ((unsigned)b < (unsigned)MAXBINS) atomicAdd(&lh[b], 1u);
    }

    atomicMax(&lmax, fkey(m));
    __syncthreads();
    for (int k = tid; k < MAXBINS; k += NTHREADS) {
        unsigned c = lh[k];
        if (c) atomicAdd(&ghist[k], c);
    }
    if (tid == 0) atomicMax(gmaxkey, lmax);
}

// ---------------------------------------------------------------------------
// Kernel 2 (1 block, 512 threads): nbins = ceil(max); normalized weights.
// jnp.histogram's last bin is right-inclusive: tar == nbins (floor == nbins)
// belongs in hist[nbins-1].
// ---------------------------------------------------------------------------
__global__ void pwl_weights_kernel(const unsigned* __restrict__ ghist,
                                   const unsigned* __restrict__ gmaxkey,
                                   long long n,
                                   float* __restrict__ gweights,
                                   int* __restrict__ pnbins) {
    __shared__ float red[MAXBINS];
    const int tid = threadIdx.x;  // 512 threads

    float mx = funkey(*gmaxkey);
    int nb = (int)ceilf(mx);
    if (nb < 1) nb = 1;
    if (nb > MAXBINS - 1) nb = MAXBINS - 1;

    unsigned h = ghist[tid];
    if (tid == nb - 1) h += ghist[nb];   // right-inclusive last bin

    float w = 0.0f;
    if (tid < nb) {
        float f = (float)h / (float)n;   // f_dems
        w = 1.0f / (f + 0.01f);          // LAMBDA_L2 == 1
    }
    red[tid] = w;
    __syncthreads();
    for (int off = MAXBINS / 2; off > 0; off >>= 1) {
        if (tid < off) red[tid] += red[tid + off];
        __syncthreads();
    }
    float s = red[0];
    gweights[tid] = (tid < nb) ? (w / s) : 0.0f;
    if (tid == 0) *pnbins = nb;
}

// ---------------------------------------------------------------------------
// Kernel 3: weighted SSE, 4-stage async global->LDS streaming pipeline.
// Dynamic LDS layout (base offset 0, no static __shared__ in this kernel):
//   [0    , 2048)  : float weights[512]
//   [2048 , 34816) : per-wave stream buffers: 8 waves x 4 stages x
//                    (512B tar | 512B src) = 8 x 4096 B
//   [34816, 34880) : 8 doubles, per-wave partials
// Each wave streams 128-float chunks; all loop control is wave-uniform so
// EXEC is all-ones when async ops issue.
// ---------------------------------------------------------------------------
__device__ __forceinline__ double pwl_term(float s, float t, const float* wl, int nb) {
    int idx = (int)floorf(t);
    float w = ((unsigned)idx < (unsigned)nb) ? wl[idx] : 0.0f;
    float d = s - t;
    return (double)(w * d * d);
}

__device__ __forceinline__ void pwl_issue_chunk(uint32_t waveBase, uint32_t laneB, int s,
                                                long long c, uint64_t tb, uint64_t sb) {
    uint32_t go = (uint32_t)(c << 9) + laneB;                    // chunk*512 bytes
    uint32_t l  = waveBase + (uint32_t)s * 1024u + laneB;
    async_ld16(l,         go, tb);                               // tar -> LDS
    async_ld16(l + 512u,  go, sb);                               // src -> LDS
}

__global__ void pwl_wsse_kernel(const float* __restrict__ src, const float* __restrict__ tar,
                                long long n,
                                const float* __restrict__ gweights,
                                const int* __restrict__ pnbins,
                                double* __restrict__ partials) {
    extern __shared__ char smem[];
    float* wl = (float*)smem;

    const int tid  = threadIdx.x;
    const int lane = tid & 31;
    const int wave = tid >> 5;

    for (int i = tid; i < MAXBINS; i += NTHREADS) wl[i] = gweights[i];
    const int nb = *pnbins;
    __syncthreads();

    const uint32_t waveBase = 2048u + (uint32_t)wave * (STAGES * 1024u);
    const uint32_t laneB    = (uint32_t)lane * 16u;
    const uint64_t tb = (uint64_t)(uintptr_t)tar;
    const uint64_t sb = (uint64_t)(uintptr_t)src;

    const long long nvec    = n >> 2;
    const long long nchunks = nvec >> 5;                        // 128 floats / chunk
    const long long tw      = (long long)gridDim.x * (NTHREADS / 32);
    const long long gw      = (long long)blockIdx.x * (NTHREADS / 32) + wave;

    double acc = 0.0;

    // prologue: issue up to STAGES-1 chunks ahead
    int issued = 0;                                             // outstanding stages
    for (int p = 0; p < STAGES - 1; ++p) {
        long long cc = gw + (long long)p * tw;
        if (cc < nchunks) { pwl_issue_chunk(waveBase, laneB, p, cc, tb, sb); ++issued; }
    }

    int stage = 0;
    long long cn_issue = gw + (long long)(STAGES - 1) * tw;
    for (long long c = gw; c < nchunks; c += tw) {
        if (cn_issue < nchunks) {
            pwl_issue_chunk(waveBase, laneB, (stage + STAGES - 1) & (STAGES - 1), cn_issue, tb, sb);
            cn_issue += tw;
            ++issued;
        }
        async_wait_allow((issued - 1) * 2);                     // oldest stage complete
        --issued;

        const uint32_t sOff = waveBase + (uint32_t)stage * 1024u + laneB;
        const float4 tv = *(const float4*)(smem + sOff);
        const float4 sv = *(const float4*)(smem + sOff + 512u);
        acc += pwl_term(sv.x, tv.x, wl, nb);
        acc += pwl_term(sv.y, tv.y, wl, nb);
        acc += pwl_term(sv.z, tv.z, wl, nb);
        acc += pwl_term(sv.w, tv.w, wl, nb);
        stage = (stage + 1) & (STAGES - 1);
    }

    // tail (elements not covered by full 128-float chunks): block 0 only
    if (blockIdx.x == 0) {
        for (long long i = nchunks * 128 + tid; i < n; i += NTHREADS)
            acc += pwl_term(src[i], tar[i], wl, nb);
    }

    // wave32 reduction, then cross-wave via LDS
    for (int off = 16; off > 0; off >>= 1) acc += __shfl_down(acc, off, 32);
    double* wpart = (double*)(smem + 2048 + 8 * (STAGES * 1024));
    if (lane == 0) wpart[wave] = acc;
    __syncthreads();
    if (tid == 0) {
        double b = 0.0;
        for (int i = 0; i < NTHREADS / 32; ++i) b += wpart[i];
        partials[blockIdx.x] = b;
    }
}

// ---------------------------------------------------------------------------
// Kernel 4: reduce block partials -> loss = sum / N (fp32 out)
// ---------------------------------------------------------------------------
__global__ void pwl_final_kernel(const double* __restrict__ partials, int nparts,
                                 long long n, float* __restrict__ out) {
    __shared__ double red[NTHREADS];
    const int tid = threadIdx.x;
    double a = 0.0;
    for (int i = tid; i < nparts; i += NTHREADS) a += partials[i];
    red[tid] = a;
    __syncthreads();
    for (int off = NTHREADS / 2; off > 0; off >>= 1) {
        if (tid < off) red[tid] += red[tid + off];
        __syncthreads();
    }
    if (tid == 0) out[0] = (float)(red[0] / (double)n);
}

// ---------------------------------------------------------------------------
extern "C" void kernel_launch(void* const* d_in, const int* in_sizes, int n_in,
                              void* d_out, int out_size, void* d_ws, size_t ws_size,
                              hipStream_t stream) {
    const float* src = (const float*)d_in[0];
    const float* tar = (const float*)d_in[1];
    const long long n = (long long)in_sizes[0];

    char* ws = (char*)d_ws;
    unsigned* ghist    = (unsigned*)(ws);            // 512*4  = 2048 B
    unsigned* gmaxkey  = (unsigned*)(ws + 2048);     // 4 B
    int*      pnbins   = (int*)     (ws + 2056);     // 4 B
    float*    gweights = (float*)   (ws + 2064);     // 512*4  = 2048 B -> 4112
    double*   partials = (double*)  (ws + 4160);     // 1024*8 = 8192 B -> 12352

    const size_t wsse_lds = 2048 + 8 * (STAGES * 1024) + 64;   // 34880 B

    pwl_init_kernel<<<1, MAXBINS, 0, stream>>>(ghist, gmaxkey);
    pwl_hist_kernel<<<HBLOCKS, NTHREADS, 0, stream>>>(tar, n, ghist, gmaxkey);
    pwl_weights_kernel<<<1, MAXBINS, 0, stream>>>(ghist, gmaxkey, n, gweights, pnbins);
    pwl_wsse_kernel<<<WBLOCKS, NTHREADS, wsse_lds, stream>>>(src, tar, n, gweights, pnbins, partials);
    pwl_final_kernel<<<1, NTHREADS, 0, stream>>>(partials, WBLOCKS, n, (float*)d_out);
}